// EventGraphSAGE_5686536700292
// MI455X (gfx1250) — compile-verified
//
#include <hip/hip_runtime.h>
#include <hip/hip_bf16.h>

typedef __attribute__((ext_vector_type(2))) float v2f;
typedef __attribute__((ext_vector_type(8))) float v8f;

#define NNODES 100000
#define NEDGES 1200000
#define DFEAT  64   // feature dim of x and h (both 64)

// ---------------------------------------------------------------------------
// Zero a contiguous float region (grid-stride).
// ---------------------------------------------------------------------------
__global__ void zero_kernel(float* __restrict__ p, size_t n) {
  size_t i = (size_t)blockIdx.x * blockDim.x + threadIdx.x;
  const size_t stride = (size_t)gridDim.x * blockDim.x;
  for (; i < n; i += stride) p[i] = 0.0f;
}

// ---------------------------------------------------------------------------
// Edge stage: for each edge e, agg[dst[e], :] += feat[src[e], :]
// (and optionally deg[dst[e]] += 1). One thread per (edge, 16-float chunk/4):
// 16 chunks of 4 floats cover the 64-wide row. float4 gather + 4 no-return
// fp32 atomics (agent scope -> global_atomic_add_f32, no CAS loop).
// ---------------------------------------------------------------------------
__device__ __forceinline__ void atomic_fadd(float* p, float v) {
  (void)__hip_atomic_fetch_add(p, v, __ATOMIC_RELAXED, __HIP_MEMORY_SCOPE_AGENT);
}

__global__ void edge_scatter_kernel(const float* __restrict__ feat,
                                    const int* __restrict__ src,
                                    const int* __restrict__ dst,
                                    float* __restrict__ agg,
                                    float* __restrict__ deg) {
  const long long idx = (long long)blockIdx.x * blockDim.x + threadIdx.x;
  const int e = (int)(idx >> 4);
  if (e >= NEDGES) return;
  const int c = ((int)idx & 15) << 2;     // float offset within the 64-wide row
  const int s = src[e];
  const int d = dst[e];
  if (deg != nullptr && (idx & 15) == 0) atomic_fadd(deg + d, 1.0f);
  const float4 v = *(const float4*)(feat + (size_t)s * DFEAT + c);
  float* o = agg + (size_t)d * DFEAT + c;
  atomic_fadd(o + 0, v.x);
  atomic_fadd(o + 1, v.y);
  atomic_fadd(o + 2, v.z);
  atomic_fadd(o + 3, v.w);
}

// ---------------------------------------------------------------------------
// Dense stage (fp32 WMMA):
//   out[i, o] = (agg[i,:] / max(deg[i],1)) . Wl[o,:] + bias[o] + xin[i,:] . Wr[o,:]
// Block = 16-node row tile; wave w owns output columns [16w, 16w+16).
// K = 64 handled as 16 chained V_WMMA_F32_16X16X4_F32 steps per matrix path,
// accumulator carried through both paths so bias/ReLU fuse at the end.
//
// ISA VGPR layouts (wave32):
//   A 16x4 f32 : lane holds A[M = lane%16][K = 2*(lane/16) + v], v=0,1
//   B 4x16 f32 : lane holds B[K = 2*(lane/16) + v][N = lane%16]
//   D 16x16 f32: VGPR v -> M = v + 8*(lane/16), N = lane%16
// Both per-lane A/B K-values are a consecutive pair -> single float2 load.
// ---------------------------------------------------------------------------
template <int DOUT, bool RELU>
__global__ void sage_mm_kernel(const float* __restrict__ agg,
                               const float* __restrict__ deg,
                               const float* __restrict__ xin,
                               const float* __restrict__ Wl,
                               const float* __restrict__ bias,
                               const float* __restrict__ Wr,
                               float* __restrict__ out) {
  const int tile   = blockIdx.x;          // 16-node row tile
  const int lane   = threadIdx.x & 31;
  const int wave   = threadIdx.x >> 5;    // output column tile
  const int laneLo = lane & 15;
  const int laneHi = lane >> 4;
  const int row    = (tile << 4) + laneLo;        // node index for A loads
  const int ocol   = (wave << 4) + laneLo;        // output col for B loads

  const float rdeg = 1.0f / fmaxf(deg[row], 1.0f);

  const float* __restrict__ arow  = agg + (size_t)row  * DFEAT;
  const float* __restrict__ xrow  = xin + (size_t)row  * DFEAT;
  const float* __restrict__ wlrow = Wl  + (size_t)ocol * DFEAT;
  const float* __restrict__ wrrow = Wr  + (size_t)ocol * DFEAT;

  v8f acc = {};

  // mean-path: (agg * rdeg) @ Wl^T
#pragma unroll
  for (int s = 0; s < DFEAT / 4; ++s) {
    const int k = (s << 2) + (laneHi << 1);
    const float2 av = *(const float2*)(arow + k);
    const float2 bv = *(const float2*)(wlrow + k);
    v2f A; A[0] = av.x * rdeg; A[1] = av.y * rdeg;
    v2f B; B[0] = bv.x;        B[1] = bv.y;
    acc = __builtin_amdgcn_wmma_f32_16x16x4_f32(
        /*neg_a=*/false, A, /*neg_b=*/false, B,
        /*c_mod=*/(short)0, acc, /*reuse_a=*/false, /*reuse_b=*/false);
  }

  // root-path: xin @ Wr^T (accumulate into same C)
#pragma unroll
  for (int s = 0; s < DFEAT / 4; ++s) {
    const int k = (s << 2) + (laneHi << 1);
    const float2 av = *(const float2*)(xrow + k);
    const float2 bv = *(const float2*)(wrrow + k);
    v2f A; A[0] = av.x; A[1] = av.y;
    v2f B; B[0] = bv.x; B[1] = bv.y;
    acc = __builtin_amdgcn_wmma_f32_16x16x4_f32(
        false, A, false, B, (short)0, acc, false, false);
  }

  const float bb = bias[ocol];
#pragma unroll
  for (int v = 0; v < 8; ++v) {
    float val = acc[v] + bb;
    if (RELU) val = fmaxf(val, 0.0f);
    const int m = (tile << 4) + (laneHi << 3) + v;   // node row of D element v
    out[(size_t)m * DOUT + ocol] = val;
  }
}

// ---------------------------------------------------------------------------
// Launch: zero -> scatter(x)+deg -> WMMA layer1 (ReLU) -> scatter(h) ->
//         WMMA layer2 -> d_out
// ---------------------------------------------------------------------------
extern "C" void kernel_launch(void* const* d_in, const int* in_sizes, int n_in,
                              void* d_out, int out_size, void* d_ws, size_t ws_size,
                              hipStream_t stream) {
  const float* x   = (const float*)d_in[0];
  const int*   ei  = (const int*)d_in[1];
  const float* W1l = (const float*)d_in[2];
  const float* b1  = (const float*)d_in[3];
  const float* W1r = (const float*)d_in[4];
  const float* W2l = (const float*)d_in[5];
  const float* b2  = (const float*)d_in[6];
  const float* W2r = (const float*)d_in[7];
  float* out = (float*)d_out;

  const int* src = ei;            // edge_index[0, :]
  const int* dst = ei + NEDGES;   // edge_index[1, :]

  // Workspace layout (floats): [deg | agg1 | agg2 | h]
  // deg+agg1+agg2 are contiguous so one zeroing pass covers all accumulators.
  float* deg  = (float*)d_ws;                               // NNODES
  float* agg1 = deg  + NNODES;                              // NNODES*64
  float* agg2 = agg1 + (size_t)NNODES * DFEAT;              // NNODES*64
  float* h    = agg2 + (size_t)NNODES * DFEAT;              // NNODES*64

  // 1) zero accumulators (deg, agg1, agg2): N*(1+64+64) floats
  {
    const size_t n = (size_t)NNODES * (1 + 2 * DFEAT);
    zero_kernel<<<4096, 256, 0, stream>>>(deg, n);
  }

  // 2) layer-1 edge stage: agg1 += x[src] scattered to dst, deg += 1
  {
    const long long threads = (long long)NEDGES * 16;
    const int blocks = (int)((threads + 255) / 256);
    edge_scatter_kernel<<<blocks, 256, 0, stream>>>(x, src, dst, agg1, deg);
  }

  // 3) layer-1 dense stage: h = relu(mean1 @ W1l^T + b1 + x @ W1r^T)
  sage_mm_kernel<64, true><<<NNODES / 16, 4 * 32, 0, stream>>>(
      agg1, deg, x, W1l, b1, W1r, h);

  // 4) layer-2 edge stage: agg2 += h[src] scattered to dst (same degrees)
  {
    const long long threads = (long long)NEDGES * 16;
    const int blocks = (int)((threads + 255) / 256);
    edge_scatter_kernel<<<blocks, 256, 0, stream>>>(h, src, dst, agg2, nullptr);
  }

  // 5) layer-2 dense stage: out = mean2 @ W2l^T + b2 + h @ W2r^T
  sage_mm_kernel<32, false><<<NNODES / 16, 2 * 32, 0, stream>>>(
      agg2, deg, h, W2l, b2, W2r, out);
}